// RouterWithLoadBalancing_66718021976459
// MI455X (gfx1250) — compile-verified
//
#include <hip/hip_runtime.h>
#include <math.h>

typedef __attribute__((ext_vector_type(2))) float v2f;
typedef __attribute__((ext_vector_type(8))) float v8f;

#define N_TOKENS      16384
#define D_MODEL       4096
#define N_EXPERTS     64
#define TOK_PER_BLOCK 128   // 8 waves * 16 tokens
#define THREADS       256

// ---------------------------------------------------------------------------
// Kernel 0: zero the workspace accumulators (64 prob-sums + 64 top1 counts)
// ---------------------------------------------------------------------------
__global__ void router_zero_ws(float* ws) {
    ws[threadIdx.x] = 0.0f;   // launched with 128 threads
}

// ---------------------------------------------------------------------------
// Kernel 1: WMMA gemm (logits = x @ gate_w^T), softmax, top-2, partial sums
// ---------------------------------------------------------------------------
__global__ __launch_bounds__(THREADS) void router_main(
    const float* __restrict__ x,      // [N_TOKENS, D_MODEL]
    const float* __restrict__ gw,     // [N_EXPERTS, D_MODEL]
    float* __restrict__ out,          // weights | indices | loss
    float* __restrict__ psum_g,       // [64] global prob sums
    float* __restrict__ cnt_g)        // [64] global top1 counts
{
    __shared__ float logits[TOK_PER_BLOCK][N_EXPERTS + 1];  // pad -> no bank conflicts
    __shared__ float psum_s[N_EXPERTS];
    __shared__ float cnt_s[N_EXPERTS];

    const int tid  = threadIdx.x;
    const int wave = tid >> 5;
    const int lane = tid & 31;
    const int lo   = lane & 15;      // column within 16-wide stripe
    const int hi   = lane >> 4;      // upper/lower half of wave

    const int tokBase = blockIdx.x * TOK_PER_BLOCK + wave * 16;

    // ---- GEMM: 16 tokens x 64 experts per wave, K in steps of 4 -----------
    // A(16x4 f32): lane holds x[tok + lo][k + 2*hi + {0,1}]
    // B(4x16 f32): vgpr r holds gw[e16 + lo][k + 2*hi + r] -> float2 load
    v8f acc0 = {}, acc1 = {}, acc2 = {}, acc3 = {};

    const float* xrow = x  + (size_t)(tokBase + lo) * D_MODEL + 2 * hi;
    const float* g0   = gw + (size_t)(lo +  0) * D_MODEL + 2 * hi;
    const float* g1   = gw + (size_t)(lo + 16) * D_MODEL + 2 * hi;
    const float* g2   = gw + (size_t)(lo + 32) * D_MODEL + 2 * hi;
    const float* g3   = gw + (size_t)(lo + 48) * D_MODEL + 2 * hi;

#pragma unroll 4
    for (int k = 0; k < D_MODEL; k += 4) {
        v2f a  = __builtin_nontemporal_load((const v2f*)(xrow + k));  // stream x (NT)
        v2f b0 = *(const v2f*)(g0 + k);                               // gate_w via cache
        v2f b1 = *(const v2f*)(g1 + k);
        v2f b2 = *(const v2f*)(g2 + k);
        v2f b3 = *(const v2f*)(g3 + k);
        acc0 = __builtin_amdgcn_wmma_f32_16x16x4_f32(false, a, false, b0, (short)0, acc0, false, false);
        acc1 = __builtin_amdgcn_wmma_f32_16x16x4_f32(false, a, false, b1, (short)0, acc1, false, false);
        acc2 = __builtin_amdgcn_wmma_f32_16x16x4_f32(false, a, false, b2, (short)0, acc2, false, false);
        acc3 = __builtin_amdgcn_wmma_f32_16x16x4_f32(false, a, false, b3, (short)0, acc3, false, false);
    }

    // C/D layout: vgpr r, lane -> (M = r + 8*hi, N = lo)
    const int mrow = wave * 16 + 8 * hi;
#pragma unroll
    for (int r = 0; r < 8; ++r) {
        logits[mrow + r][lo +  0] = acc0[r];
        logits[mrow + r][lo + 16] = acc1[r];
        logits[mrow + r][lo + 32] = acc2[r];
        logits[mrow + r][lo + 48] = acc3[r];
    }

    if (tid < N_EXPERTS) { psum_s[tid] = 0.0f; cnt_s[tid] = 0.0f; }
    __syncthreads();

    // ---- per-token softmax / top-2 / partial reductions -------------------
    if (tid < TOK_PER_BLOCK) {
        const int t = tid;

        // stable top-2 (strict '>' keeps lowest index on ties, like lax.top_k)
        float v1 = -INFINITY, v2 = -INFINITY;
        int   i1 = 0,         i2 = 0;
#pragma unroll 8
        for (int e = 0; e < N_EXPERTS; ++e) {
            float l = logits[t][e];
            if (l > v1)      { v2 = v1; i2 = i1; v1 = l; i1 = e; }
            else if (l > v2) { v2 = l;  i2 = e; }
        }

        // softmax denominator (stable around v1 = row max)
        float s = 0.0f;
#pragma unroll 8
        for (int e = 0; e < N_EXPERTS; ++e)
            s += __expf(logits[t][e] - v1);
        const float inv = 1.0f / s;

        // per-expert prob sums, staggered start to spread LDS atomics
        for (int eo = 0; eo < N_EXPERTS; ++eo) {
            int e = (eo + t) & (N_EXPERTS - 1);
            atomicAdd(&psum_s[e], __expf(logits[t][e] - v1) * inv);
        }
        atomicAdd(&cnt_s[i1], 1.0f);

        // renormalized top-2 routing weights: [1/(1+e2), e2/(1+e2)]
        const float e2    = __expf(v2 - v1);
        const float denom = 1.0f + e2;
        const int   g     = blockIdx.x * TOK_PER_BLOCK + t;
        out[2 * g + 0] = 1.0f / denom;
        out[2 * g + 1] = e2 / denom;
        out[2 * N_TOKENS + 2 * g + 0] = (float)i1;   // indices flattened as f32
        out[2 * N_TOKENS + 2 * g + 1] = (float)i2;
    }
    __syncthreads();

    if (tid < N_EXPERTS) {
        atomicAdd(&psum_g[tid], psum_s[tid]);
        atomicAdd(&cnt_g[tid],  cnt_s[tid]);
    }
}

// ---------------------------------------------------------------------------
// Kernel 2: loss = 0.01 * sum_e (cnt_e/N) * (psum_e/N)
// ---------------------------------------------------------------------------
__global__ void router_finalize(const float* __restrict__ psum,
                                const float* __restrict__ cnt,
                                float* __restrict__ out) {
    __shared__ float red[N_EXPERTS];
    const int e = threadIdx.x;
    red[e] = psum[e] * cnt[e];
    __syncthreads();
    if (e == 0) {
        float s = 0.0f;
#pragma unroll
        for (int i = 0; i < N_EXPERTS; ++i) s += red[i];
        out[4 * N_TOKENS] = 0.01f * s / ((float)N_TOKENS * (float)N_TOKENS);
    }
}

// ---------------------------------------------------------------------------
extern "C" void kernel_launch(void* const* d_in, const int* in_sizes, int n_in,
                              void* d_out, int out_size, void* d_ws, size_t ws_size,
                              hipStream_t stream) {
    const float* x  = (const float*)d_in[0];   // [16384, 4096] f32
    const float* gw = (const float*)d_in[1];   // [64, 4096] f32
    float* out  = (float*)d_out;
    float* psum = (float*)d_ws;                // 64 floats
    float* cnt  = psum + N_EXPERTS;            // 64 floats

    router_zero_ws<<<1, 2 * N_EXPERTS, 0, stream>>>(psum);
    router_main<<<N_TOKENS / TOK_PER_BLOCK, THREADS, 0, stream>>>(x, gw, out, psum, cnt);
    router_finalize<<<1, N_EXPERTS, 0, stream>>>(psum, cnt, out);
}